// RnnAC_65283502899501
// MI455X (gfx1250) — compile-verified
//
#include <hip/hip_runtime.h>
#include <math.h>

// ---------------------------------------------------------------------------
// 2-layer LSTM (H=8) actor-critic head, CDNA5 (gfx1250), wave32.
// One wave processes a 16-row batch tile for all 2048 timesteps.
// Gate GEMM per layer/step:  [16 x 16] x [16 x 32]  via 8x V_WMMA_F32_16X16X4_F32
// (4 K-steps x 2 N-tiles), f32 end-to-end to match the f32 reference.
// h-state transposed back into A-fragment layout through wave-private LDS.
// x_t fragments software-pipelined with a structural (manually unrolled x2)
// double buffer -> no dynamic register indexing.
// Activations: activate-own-then-shuffle; 3 exp + 3 rcp per 8-gate group.
// ---------------------------------------------------------------------------

typedef float v2f __attribute__((ext_vector_type(2)));
typedef float v8f __attribute__((ext_vector_type(8)));

#define B_TOT 4096
#define S_LEN 2048
#define IN_DIM 8
#define H_DIM 8
#define LIN_DIM 4
#define A_DIM 4

#define TILE_M 16
#define WAVES_PER_BLOCK 4

__device__ __forceinline__ float fast_sigmoid(float x) {
    return __builtin_amdgcn_rcpf(1.0f + __expf(-x));
}
__device__ __forceinline__ float fast_tanh(float x) {
    return fmaf(2.0f, __builtin_amdgcn_rcpf(1.0f + __expf(-2.0f * x)), -1.0f);
}

__global__ __launch_bounds__(WAVES_PER_BLOCK * 32)
void RnnAC_lstm_wmma_kernel(
    const float* __restrict__ seq, const int* __restrict__ lengths,
    const float* __restrict__ Wih0, const float* __restrict__ Whh0,
    const float* __restrict__ bih0, const float* __restrict__ bhh0,
    const float* __restrict__ Wih1, const float* __restrict__ Whh1,
    const float* __restrict__ bih1, const float* __restrict__ bhh1,
    const float* __restrict__ Wl,   const float* __restrict__ bl,
    const float* __restrict__ Wv,   const float* __restrict__ bv,
    const float* __restrict__ Wa,   const float* __restrict__ ba,
    const float* __restrict__ log_std,
    float* __restrict__ out)
{
    // wave-private h transpose staging: 2 buffers of 16x8 f32 per wave
    __shared__ float lds[WAVES_PER_BLOCK * 2 * TILE_M * H_DIM];

    const int lane  = threadIdx.x & 31;
    const int wave  = threadIdx.x >> 5;
    const int bbase = (blockIdx.x * WAVES_PER_BLOCK + wave) * TILE_M;

    float* hbuf0 = &lds[wave * 2 * TILE_M * H_DIM];
    float* hbuf1 = hbuf0 + TILE_M * H_DIM;

    const int  nlo = lane & 15;        // N (B/C/D frags) or M (A frags)
    const int  hi2 = (lane >> 4) & 1;  // upper 16-lane half
    const int  n8  = lane & 7;         // hidden index in lane-state
    const bool low = ((lane & 8) == 0);

    // tile1 (g|o) activation constants: low lanes own g (tanh), high own o (sigmoid)
    // act = ca * rcp(1 + exp(es*x)) + cb
    const float es = low ? -2.0f : -1.0f;
    const float ca = low ?  2.0f :  1.0f;
    const float cb = low ? -1.0f :  0.0f;

    // ---- weight B-fragments, resident for whole loop ----------------------
    // W-hat[16K x 32N]: k<8 -> Wih.T (Wih[N][k]), k>=8 -> Whh.T (Whh[N][k-8])
    // B frag layout (16x16x4): VGPR v, lanes: k = kb + v + 2*hi2, N = nt*16+nlo
    v2f Bw0[4][2], Bw1[4][2];
#pragma unroll
    for (int kb4 = 0; kb4 < 4; ++kb4) {
#pragma unroll
        for (int nt = 0; nt < 2; ++nt) {
#pragma unroll
            for (int v = 0; v < 2; ++v) {
                int k = kb4 * 4 + v + 2 * hi2;
                int N = nt * 16 + nlo;
                float w0, w1;
                if (k < 8) { w0 = Wih0[N * IN_DIM + k];       w1 = Wih1[N * H_DIM + k]; }
                else       { w0 = Whh0[N * H_DIM + (k - 8)];  w1 = Whh1[N * H_DIM + (k - 8)]; }
                Bw0[kb4][nt][v] = w0;
                Bw1[kb4][nt][v] = w1;
            }
        }
    }
    // bias C-vectors (depend only on N); used directly as WMMA C operands
    v8f bias00, bias01, bias10, bias11;
#pragma unroll
    for (int i = 0; i < 8; ++i) {
        bias00[i] = bih0[nlo] + bhh0[nlo];
        bias01[i] = bih0[16 + nlo] + bhh0[16 + nlo];
        bias10[i] = bih1[nlo] + bhh1[nlo];
        bias11[i] = bih1[16 + nlo] + bhh1[16 + nlo];
    }

    // per-lane per-register sequence lengths (m = r + 8*hi2)
    int lenr[8];
#pragma unroll
    for (int r = 0; r < 8; ++r) lenr[r] = lengths[bbase + r + hi2 * 8];

    // zero the h staging buffers (wave-private, no barrier needed)
    for (int i = lane; i < 2 * TILE_M * H_DIM; i += 32) hbuf0[i] = 0.0f;

    // lane-state: element (m = r + 8*hi2, n = n8), duplicated across lane^8
    float c0[8], c1[8], y[8];
#pragma unroll
    for (int r = 0; r < 8; ++r) { c0[r] = 0.f; c1[r] = 0.f; y[r] = 0.f; }

    const float* xrow = seq + (size_t)(bbase + nlo) * (S_LEN * IN_DIM);

    // load x A-fragments (K0..7 of layer 0) for timestep t
    auto loadx = [&](v2f Axd[2], int t) {
        const float* xp = xrow + t * IN_DIM;
        Axd[0][0] = xp[0 + 2 * hi2];
        Axd[0][1] = xp[1 + 2 * hi2];
        Axd[1][0] = xp[4 + 2 * hi2];
        Axd[1][1] = xp[5 + 2 * hi2];
    };

    // one full LSTM timestep (both layers) using pre-loaded x fragments
    auto lstm_step = [&](int t, const v2f Axc[2]) {
        // ---- layer 0: A = [x_t | h0(prev via LDS)] ----
        v2f Ah0[2];
        Ah0[0][0] = hbuf0[nlo * 8 + 0 + 2 * hi2];
        Ah0[0][1] = hbuf0[nlo * 8 + 1 + 2 * hi2];
        Ah0[1][0] = hbuf0[nlo * 8 + 4 + 2 * hi2];
        Ah0[1][1] = hbuf0[nlo * 8 + 5 + 2 * hi2];

        v8f G0 = bias00, G1 = bias01;
        G0 = __builtin_amdgcn_wmma_f32_16x16x4_f32(false, Axc[0], false, Bw0[0][0], (short)0, G0, false, false);
        G1 = __builtin_amdgcn_wmma_f32_16x16x4_f32(false, Axc[0], false, Bw0[0][1], (short)0, G1, false, false);
        G0 = __builtin_amdgcn_wmma_f32_16x16x4_f32(false, Axc[1], false, Bw0[1][0], (short)0, G0, false, false);
        G1 = __builtin_amdgcn_wmma_f32_16x16x4_f32(false, Axc[1], false, Bw0[1][1], (short)0, G1, false, false);
        G0 = __builtin_amdgcn_wmma_f32_16x16x4_f32(false, Ah0[0], false, Bw0[2][0], (short)0, G0, false, false);
        G1 = __builtin_amdgcn_wmma_f32_16x16x4_f32(false, Ah0[0], false, Bw0[2][1], (short)0, G1, false, false);
        G0 = __builtin_amdgcn_wmma_f32_16x16x4_f32(false, Ah0[1], false, Bw0[3][0], (short)0, G0, false, false);
        G1 = __builtin_amdgcn_wmma_f32_16x16x4_f32(false, Ah0[1], false, Bw0[3][1], (short)0, G1, false, false);

        float h0new[8];
#pragma unroll
        for (int r = 0; r < 8; ++r) {
            float s0 = fast_sigmoid(G0[r]);                    // own i or f
            float q0 = __shfl_xor(s0, 8, 32);
            float gi = low ? s0 : q0;
            float gf = low ? q0 : s0;
            float s1 = fmaf(ca, __builtin_amdgcn_rcpf(1.0f + __expf(es * G1[r])), cb);
            float q1 = __shfl_xor(s1, 8, 32);
            float gg = low ? s1 : q1;
            float go = low ? q1 : s1;
            float c = fmaf(gf, c0[r], gi * gg);
            c0[r] = c;
            h0new[r] = go * fast_tanh(c);
        }
        if (low) {
#pragma unroll
            for (int r = 0; r < 8; ++r)
                hbuf0[(r + hi2 * 8) * 8 + n8] = h0new[r];
        }

        // ---- layer 1: A = [h0(new) | h1(prev)] both via LDS ----
        v2f A1[4];
        A1[0][0] = hbuf0[nlo * 8 + 0 + 2 * hi2];
        A1[0][1] = hbuf0[nlo * 8 + 1 + 2 * hi2];
        A1[1][0] = hbuf0[nlo * 8 + 4 + 2 * hi2];
        A1[1][1] = hbuf0[nlo * 8 + 5 + 2 * hi2];
        A1[2][0] = hbuf1[nlo * 8 + 0 + 2 * hi2];
        A1[2][1] = hbuf1[nlo * 8 + 1 + 2 * hi2];
        A1[3][0] = hbuf1[nlo * 8 + 4 + 2 * hi2];
        A1[3][1] = hbuf1[nlo * 8 + 5 + 2 * hi2];

        v8f H0 = bias10, H1 = bias11;
#pragma unroll
        for (int kb = 0; kb < 4; ++kb) {
            H0 = __builtin_amdgcn_wmma_f32_16x16x4_f32(false, A1[kb], false, Bw1[kb][0], (short)0, H0, false, false);
            H1 = __builtin_amdgcn_wmma_f32_16x16x4_f32(false, A1[kb], false, Bw1[kb][1], (short)0, H1, false, false);
        }
        float h1new[8];
#pragma unroll
        for (int r = 0; r < 8; ++r) {
            float s0 = fast_sigmoid(H0[r]);
            float q0 = __shfl_xor(s0, 8, 32);
            float gi = low ? s0 : q0;
            float gf = low ? q0 : s0;
            float s1 = fmaf(ca, __builtin_amdgcn_rcpf(1.0f + __expf(es * H1[r])), cb);
            float q1 = __shfl_xor(s1, 8, 32);
            float gg = low ? s1 : q1;
            float go = low ? q1 : s1;
            float c = fmaf(gf, c1[r], gi * gg);
            c1[r] = c;
            h1new[r] = go * fast_tanh(c);
            // packed-sequence gather: keep output at t == length-1
            y[r] = (t == lenr[r] - 1) ? h1new[r] : y[r];
        }
        if (low) {
#pragma unroll
            for (int r = 0; r < 8; ++r)
                hbuf1[(r + hi2 * 8) * 8 + n8] = h1new[r];
        }
    };

    // ---- time loop, manually unrolled x2 with structural double buffer ----
    v2f Axa[2], Axb[2];
    loadx(Axa, 0);
    for (int t = 0; t < S_LEN; t += 2) {
        // stage t+1's x while computing step t
        loadx(Axb, t + 1);
        {
            const int tp = (t + 8 < S_LEN) ? (t + 8) : t;
            __builtin_prefetch(xrow + tp * IN_DIM, 0, 3);
        }
        lstm_step(t, Axa);
        // stage t+2's x while computing step t+1 (clamped on last iteration)
        loadx(Axa, (t + 2 < S_LEN) ? (t + 2) : (t + 1));
        lstm_step(t + 1, Axb);
    }

    // ---- heads: feat = tanh(y @ Wl.T + bl); value / action heads ----------
    if (low) {
#pragma unroll
        for (int r = 0; r < 8; ++r)
            hbuf0[(r + hi2 * 8) * 8 + n8] = y[r];
    }
    if (lane < 16) {
        int row = bbase + lane;
        float yv[8];
#pragma unroll
        for (int n = 0; n < 8; ++n) yv[n] = hbuf0[lane * 8 + n];
        float feat[LIN_DIM];
#pragma unroll
        for (int j = 0; j < LIN_DIM; ++j) {
            float acc = bl[j];
#pragma unroll
            for (int n = 0; n < 8; ++n) acc = fmaf(Wl[j * H_DIM + n], yv[n], acc);
            feat[j] = fast_tanh(acc);
        }
        float val = bv[0];
#pragma unroll
        for (int j = 0; j < LIN_DIM; ++j) val = fmaf(Wv[j], feat[j], val);
        out[row] = val;                                     // value   [0, B)
#pragma unroll
        for (int a = 0; a < A_DIM; ++a) {
            float am = ba[a];
#pragma unroll
            for (int j = 0; j < LIN_DIM; ++j) am = fmaf(Wa[a * LIN_DIM + j], feat[j], am);
            float ls = log_std[a];
            out[B_TOT + row * A_DIM + a]     = am;          // action_mean
            out[5 * B_TOT + row * A_DIM + a] = ls;          // action_log_std
            out[9 * B_TOT + row * A_DIM + a] = __expf(ls);  // action_std
        }
    }
}

extern "C" void kernel_launch(void* const* d_in, const int* in_sizes, int n_in,
                              void* d_out, int out_size, void* d_ws, size_t ws_size,
                              hipStream_t stream) {
    const float* seq     = (const float*)d_in[0];
    const int*   lengths = (const int*)d_in[1];
    const float* Wih0    = (const float*)d_in[2];
    const float* Whh0    = (const float*)d_in[3];
    const float* bih0    = (const float*)d_in[4];
    const float* bhh0    = (const float*)d_in[5];
    const float* Wih1    = (const float*)d_in[6];
    const float* Whh1    = (const float*)d_in[7];
    const float* bih1    = (const float*)d_in[8];
    const float* bhh1    = (const float*)d_in[9];
    const float* Wl      = (const float*)d_in[10];
    const float* bl      = (const float*)d_in[11];
    const float* Wv      = (const float*)d_in[12];
    const float* bv      = (const float*)d_in[13];
    const float* Wa      = (const float*)d_in[14];
    const float* ba      = (const float*)d_in[15];
    const float* log_std = (const float*)d_in[16];

    const int blocks = B_TOT / (TILE_M * WAVES_PER_BLOCK);   // 64
    RnnAC_lstm_wmma_kernel<<<blocks, WAVES_PER_BLOCK * 32, 0, stream>>>(
        seq, lengths, Wih0, Whh0, bih0, bhh0, Wih1, Whh1, bih1, bhh1,
        Wl, bl, Wv, bv, Wa, ba, log_std, (float*)d_out);
}